// SparseAffinity_Propagate_38311108280732
// MI455X (gfx1250) — compile-verified
//
#include <hip/hip_runtime.h>
#include <hip/hip_bf16.h>
#include <stdint.h>

// ---------------------------------------------------------------------------
// Sparse affinity propagation (SPN), MI455X / gfx1250.
//
// Folded form: per pixel, 8 f16 affinities w'_c and one f32 base term are
// iteration-invariant:  result = base + sum_c w'_c * result(p + off_c)
// Working set (75.5MB f16 weights + 19MB base + 2x19.5MB padded ping-pong
// buffers) ~= 133MB -> resident in the 192MB L2, so the 24-iteration loop is
// L2-bandwidth bound rather than HBM bound (23.3 TB/s would otherwise see
// ~4.6GB of traffic with f32 weights).
//
// CDNA5 specifics used:
//  - global_load_async_to_lds_b128 + s_wait_asynccnt for halo-tile staging
//  - th:TH_STORE_WB on result stores (keep ping-pong data dirty in L2,
//    override the default wr-rinse policy)
// ---------------------------------------------------------------------------

#define HH   768
#define WW   768
#define BB   8
#define PAD  5
#define PITCH 784                 // padded row pitch (floats), 16B-aligned rows
#define ROWS  (HH + 2*PAD)        // 778
#define IMG   (ROWS * PITCH)      // elems per padded image (609,952; %4 == 0)

#define TX   64                   // tile pixels in x
#define TY   64                   // tile pixels in y
#define LW   76                   // LDS row width (need 74, pad to 76 = 19*4)
#define LH   (TY + 2*PAD)         // 74
#define NV4  (LH * (LW/4))        // 1406 float4 async loads per tile

typedef _Float16 h8 __attribute__((ext_vector_type(8)));

// Neighbor offsets (dy,dx), reference OFFSETS order.
#define DY0 (-5)
#define DY1 (-1)
#define DY2 ( 0)
#define DY3 ( 0)
#define DY4 ( 5)
#define DY5 ( 1)
#define DY6 ( 0)
#define DY7 ( 0)
#define DX0 ( 0)
#define DX1 ( 0)
#define DX2 ( 5)
#define DX3 ( 1)
#define DX4 ( 0)
#define DX5 ( 0)
#define DX6 (-5)
#define DX7 (-1)

// Store with WB temporal hint: keep the line dirty in far cache (L2) instead
// of the default wr-rinse. Ping-pong buffers are re-read next iteration.
__device__ __forceinline__ void store_f32_wb(float* base, unsigned byteOff, float v) {
    asm volatile("global_store_b32 %0, %1, %2 th:TH_STORE_WB"
                 :: "v"(byteOff), "v"(v), "s"(base)
                 : "memory");
}

// ---------------------------------------------------------------------------
__global__ __launch_bounds__(256) void spn_zero(float4* __restrict__ p, int n4) {
    int i = blockIdx.x * 256 + threadIdx.x;
    if (i < n4) p[i] = make_float4(0.f, 0.f, 0.f, 0.f);
}

// ---------------------------------------------------------------------------
// Precompute: normalized affinities (packed f16), base term (f32), and seed
// result0 (padded) with blur_depth.
__global__ __launch_bounds__(256) void spn_precompute(
    const float* __restrict__ guid,    // (B,8,H,W)
    const float* __restrict__ blur,    // (B,1,H,W)
    const float* __restrict__ sparse,  // (B,1,H,W)
    h8*          __restrict__ wpk,     // (B*H*W) packed 8 x f16
    float*       __restrict__ basep,   // (B*H*W)
    float*       __restrict__ buf0)    // padded result buffer 0
{
    const size_t i = (size_t)blockIdx.x * 256 + threadIdx.x;
    const int x = (int)(i % WW);
    const int y = (int)((i / WW) % HH);
    const int b = (int)(i / ((size_t)WW * HH));

    const int DYv[8] = {DY0,DY1,DY2,DY3,DY4,DY5,DY6,DY7};
    const int DXv[8] = {DX0,DX1,DX2,DX3,DX4,DX5,DX6,DX7};

    float g[8];
    float absw = 0.f;
#pragma unroll
    for (int c = 0; c < 8; ++c) {
        const int yy = y + DYv[c];
        const int xx = x + DXv[c];
        float v = 0.f;
        if (yy >= 0 && yy < HH && xx >= 0 && xx < WW)
            v = guid[(((size_t)b * 8 + c) * HH + yy) * WW + xx];
        g[c] = v;
        absw += fabsf(v);
    }
    const float inv = 1.0f / fmaxf(absw, 1e-6f);
    float gs = 0.f;
#pragma unroll
    for (int c = 0; c < 8; ++c) { g[c] *= inv; gs += g[c]; }

    const float raw  = blur[i];
    const float sp   = sparse[i];
    const bool  m    = (sp > 0.0f);               // sign(sparse) for sparse>=0
    const float base = m ? raw : (1.0f - gs) * raw;

    h8 w;
#pragma unroll
    for (int c = 0; c < 8; ++c)
        w[c] = m ? (_Float16)0.0f : (_Float16)g[c];

    wpk[i]   = w;
    basep[i] = base;
    buf0[(size_t)b * IMG + (size_t)(y + PAD) * PITCH + (x + PAD)] = raw;
}

// ---------------------------------------------------------------------------
// One propagation step. src is a padded buffer; dst is either a padded buffer
// (dstPad=PAD) or the final output (dstPad=0, pitch=W).
__global__ __launch_bounds__(256) void spn_iterate(
    const float* __restrict__ src,
    const h8*    __restrict__ wpk,
    const float* __restrict__ basep,
    float*       __restrict__ dst,
    int dstPitch, int dstImg, int dstPad)
{
    // 74*76*4 = 22,496 B of the 320KB WGP LDS; 16B aligned for b128 async.
    __shared__ __align__(16) float tile[LH * LW];

    const int tid = threadIdx.x;
    const int gx0 = blockIdx.x * TX;
    const int gy0 = blockIdx.y * TY;
    const int b   = blockIdx.z;

    // Padded coords: halo origin row = gy0, col = gx0 (pixel (y,x) lives at
    // padded (y+PAD, x+PAD); halo starts at y-PAD -> padded y). 76-wide b128
    // rows stay inside the 784-float pitch for every tile (max col 779 < 784).
    const float* sb = src + (size_t)b * IMG + (size_t)gy0 * PITCH + gx0;

    // --- async tensor-path staging: global -> LDS, 128b per lane -----------
    for (int i = tid; i < NV4; i += 256) {
        const int r  = i / (LW / 4);
        const int c4 = (i % (LW / 4)) * 4;
        unsigned lds  = (unsigned)(unsigned long long)(uintptr_t)&tile[r * LW + c4];
        unsigned voff = (unsigned)((r * PITCH + c4) * 4);
        asm volatile("global_load_async_to_lds_b128 %0, %1, %2"
                     :: "v"(lds), "v"(voff), "s"(sb)
                     : "memory");
    }
    asm volatile("s_wait_asynccnt 0x0" ::: "memory");
    __syncthreads();

    // --- compute: 16 pixels per thread --------------------------------------
    const int lx  = tid & (TX - 1);
    const int ty0 = tid >> 6;            // 0..3
    float* dstB = dst + (size_t)b * dstImg;

#pragma unroll 4
    for (int j = 0; j < TY / 4; ++j) {
        const int ly = ty0 * (TY / 4) + j;
        const int gy = gy0 + ly;
        const int gx = gx0 + lx;
        const size_t pix = ((size_t)b * HH + gy) * WW + gx;

        const h8   w   = wpk[pix];       // one b128 load, 8 x f16 weights
        float      acc = basep[pix];
        const float* t = &tile[(ly + PAD) * LW + (lx + PAD)];

        acc += (float)w[0] * t[DY0 * LW + DX0];
        acc += (float)w[1] * t[DY1 * LW + DX1];
        acc += (float)w[2] * t[DY2 * LW + DX2];
        acc += (float)w[3] * t[DY3 * LW + DX3];
        acc += (float)w[4] * t[DY4 * LW + DX4];
        acc += (float)w[5] * t[DY5 * LW + DX5];
        acc += (float)w[6] * t[DY6 * LW + DX6];
        acc += (float)w[7] * t[DY7 * LW + DX7];

        const unsigned byteOff =
            (unsigned)(((gy + dstPad) * dstPitch + (gx + dstPad)) * 4);
        store_f32_wb(dstB, byteOff, acc);
    }
}

// ---------------------------------------------------------------------------
extern "C" void kernel_launch(void* const* d_in, const int* in_sizes, int n_in,
                              void* d_out, int out_size, void* d_ws, size_t ws_size,
                              hipStream_t stream) {
    (void)in_sizes; (void)n_in; (void)out_size; (void)ws_size;

    const float* guidance = (const float*)d_in[0];   // (8,8,768,768) f32
    const float* blur     = (const float*)d_in[1];   // (8,1,768,768) f32
    const float* sparse   = (const float*)d_in[2];   // (8,1,768,768) f32
    float*       out      = (float*)d_out;           // (8,1,768,768) f32

    const size_t N = (size_t)BB * HH * WW;           // 4,718,592 pixels

    // Workspace layout (all 16B aligned):
    //   buf0: BB*IMG f32   (19.5 MB)
    //   buf1: BB*IMG f32   (19.5 MB)
    //   wpk : N * 16 B     (75.5 MB)  packed f16 weights
    //   base: N * 4 B      (18.9 MB)     total ~133.4 MB
    char*  ws    = (char*)d_ws;
    float* buf0  = (float*)ws;
    float* buf1  = buf0 + (size_t)BB * IMG;
    h8*    wpk   = (h8*)(ws + 2 * (size_t)BB * IMG * sizeof(float));
    float* basep = (float*)((char*)wpk + N * sizeof(h8));

    // 1) zero both padded buffers (halo correctness; d_ws is poisoned).
    const int n4 = (int)((2 * (size_t)BB * IMG) / 4);
    spn_zero<<<(n4 + 255) / 256, 256, 0, stream>>>((float4*)buf0, n4);

    // 2) fold affinities/mask/raw into per-pixel {8 x f16 weights, f32 base},
    //    seed buf0 with blur_depth.
    spn_precompute<<<(unsigned)(N / 256), 256, 0, stream>>>(
        guidance, blur, sparse, wpk, basep, buf0);

    // 3) 24 propagation steps, ping-pong; last one writes d_out directly.
    const dim3 grid(WW / TX, HH / TY, BB);
    for (int k = 0; k < 24; ++k) {
        const float* s = (k & 1) ? buf1 : buf0;
        float* d;
        int pitch, img, pad;
        if (k == 23) { d = out;                   pitch = WW;    img = WW * HH; pad = 0;   }
        else         { d = (k & 1) ? buf0 : buf1; pitch = PITCH; img = IMG;     pad = PAD; }
        spn_iterate<<<grid, 256, 0, stream>>>(s, wpk, basep, d, pitch, img, pad);
    }
}